// WeightedSpatialInterpolation_30408368456279
// MI455X (gfx1250) — compile-verified
//
#include <hip/hip_runtime.h>

// Problem constants (from reference)
#define B_   4
#define N_   8192
#define M_   4096
#define C1_  128
#define C2_  256
#define CO_  (C1_ + C2_)   // 384 output channels
#define EPS_ 1e-6f

typedef __attribute__((ext_vector_type(2))) float v2f;
typedef __attribute__((ext_vector_type(8))) float v8f;
typedef int v4i_ __attribute__((vector_size(16)));   // builtin's pointee type

#if __has_builtin(__builtin_amdgcn_global_load_async_to_lds_b128)
#define HAVE_ASYNC_LDS 1
#else
#define HAVE_ASYNC_LDS 0
#endif

// Branchless sorted insert into a 3-element min-list (lowers to v_cndmask).
__device__ __forceinline__ void insert3(float d, int m,
                                        float& d0, float& d1, float& d2,
                                        int& i0, int& i1, int& i2) {
    const bool l0 = d < d0;
    const bool l1 = d < d1;
    const bool l2 = d < d2;
    const float nd2 = l1 ? d1 : (l2 ? d : d2);
    const int   ni2 = l1 ? i1 : (l2 ? m : i2);
    const float nd1 = l0 ? d0 : (l1 ? d : d1);
    const int   ni1 = l0 ? i0 : (l1 ? m : i1);
    d0 = l0 ? d : d0;  i0 = l0 ? m : i0;
    d1 = nd1;          i1 = ni1;
    d2 = nd2;          i2 = ni2;
}

// ---------------------------------------------------------------------------
// Phase 1: brute-force 3-NN via V_WMMA_F32_16X16X4_F32.
//   A (16x4) row m: [-2x_s, -2y_s, -2z_s, |s|^2], B (4x16) col n: [x_t,y_t,z_t,1]
//   => acc = |s|^2 - 2 s.t ; distance^2 = acc + |t|^2.  The |s|^2 term rides in
//   the otherwise-unused K=3 slot, so the inner loop is just ds_load_b64 + wmma
//   + min-tree + (rare) branchless insert.
// A-operand staged in LDS pre-swizzled into the ISA 7.12.2 A-register layout:
//   lanes 0-15 hold K={0,1} of point m=lane, lanes 16-31 hold K={2,3}.
// ---------------------------------------------------------------------------
__global__ __launch_bounds__(256) void knn3_wmma_kernel(
    const float* __restrict__ tgt,      // [B, N, 3]
    const float* __restrict__ src,      // [B, M, 3]
    int*  __restrict__ knn_idx,         // [B, N, 3]
    float* __restrict__ knn_w)          // [B, N, 3] (normalized weights)
{
    __shared__ v2f alay[M_ * 2];        // A-layout operand pairs: 64 KB
#if HAVE_ASYNC_LDS
    __shared__ __align__(16) float raw[M_ * 3];   // raw coords: 48 KB
#endif

    const int b       = blockIdx.x >> 6;          // N_/128 == 64 blocks per batch
    const int n_block = (blockIdx.x & 63) * 128;  // 128 targets per block
    const int tid     = threadIdx.x;
    const float* gsrc = src + (size_t)b * M_ * 3;

#if HAVE_ASYNC_LDS
    // gfx1250 async copy: 48 KB global -> LDS, tracked by ASYNCcnt.
    for (int j = tid; j < (M_ * 3) / 4; j += 256) {
        __builtin_amdgcn_global_load_async_to_lds_b128(
            (v4i_*)(gsrc + 4 * j), (v4i_*)(&raw[4 * j]), 0, 0);
    }
#if __has_builtin(__builtin_amdgcn_s_wait_asynccnt)
    __builtin_amdgcn_s_wait_asynccnt(0);
#else
    asm volatile("s_wait_asynccnt 0x0" ::: "memory");
#endif
    __syncthreads();
#endif

    // Transform into WMMA A-register layout (pre-scaled by -2, |s|^2 in K=3).
    for (int m = tid; m < M_; m += 256) {
#if HAVE_ASYNC_LDS
        const float x = raw[m * 3 + 0];
        const float y = raw[m * 3 + 1];
        const float z = raw[m * 3 + 2];
#else
        const float x = gsrc[m * 3 + 0];
        const float y = gsrc[m * 3 + 1];
        const float z = gsrc[m * 3 + 2];
#endif
        const int chunk = m >> 4, pos = m & 15;
        v2f lo; lo.x = -2.0f * x; lo.y = -2.0f * y;
        v2f hv; hv.x = -2.0f * z; hv.y = x * x + y * y + z * z;
        alay[chunk * 32 + pos]      = lo;   // lanes 0-15: K={0,1}
        alay[chunk * 32 + 16 + pos] = hv;   // lanes 16-31: K={2,3}
    }
    __syncthreads();

    const int wave = tid >> 5;       // 8 waves per block
    const int lane = tid & 31;       // wave32
    const int hi   = lane >> 4;
    const int col  = lane & 15;
    const int n    = n_block + wave * 16 + col;

    // B matrix 4x16 (K x N): K rows (x, y, z, 1).
    const float tx = tgt[(b * N_ + n) * 3 + 0];
    const float ty = tgt[(b * N_ + n) * 3 + 1];
    const float tz = tgt[(b * N_ + n) * 3 + 2];
    const float t2 = tx * tx + ty * ty + tz * tz;
    v2f bmat;
    bmat.x = hi ? tz   : tx;
    bmat.y = hi ? 1.0f : ty;

    float d0 = 1e30f, d1 = 1e30f, d2 = 1e30f;
    int   i0 = 0,     i1 = 0,     i2 = 0;

    // Two 16-source tiles per iteration: 2 WMMAs, one wave-uniform reject.
    for (int m0 = 0; m0 < M_; m0 += 32) {
        const v2f amatA = alay[2 * m0 + lane];
        const v2f amatB = alay[2 * (m0 + 16) + lane];

        v8f accA = {};
        accA = __builtin_amdgcn_wmma_f32_16x16x4_f32(
                   false, amatA, false, bmat, (short)0, accA, false, false);
        v8f accB = {};
        accB = __builtin_amdgcn_wmma_f32_16x16x4_f32(
                   false, amatB, false, bmat, (short)0, accB, false, false);

        float mn = fminf(accA[0], accB[0]);
#pragma unroll
        for (int v = 1; v < 8; ++v) mn = fminf(mn, fminf(accA[v], accB[v]));

        if (__any((mn + t2) < d2)) {        // wave-uniform s_cbranch skip
#pragma unroll
            for (int v = 0; v < 8; ++v) {
                insert3(accA[v] + t2, m0 + hi * 8 + v,      d0, d1, d2, i0, i1, i2);
                insert3(accB[v] + t2, m0 + 16 + hi * 8 + v, d0, d1, d2, i0, i1, i2);
            }
        }
    }

    // Merge the two lane halves of each target column (snapshot partner first).
    const float pd0 = __shfl_xor(d0, 16, 32);
    const float pd1 = __shfl_xor(d1, 16, 32);
    const float pd2 = __shfl_xor(d2, 16, 32);
    const int   pi0 = __shfl_xor(i0, 16, 32);
    const int   pi1 = __shfl_xor(i1, 16, 32);
    const int   pi2 = __shfl_xor(i2, 16, 32);
    insert3(pd0, pi0, d0, d1, d2, i0, i1, i2);
    insert3(pd1, pi1, d0, d1, d2, i0, i1, i2);
    insert3(pd2, pi2, d0, d1, d2, i0, i1, i2);

    if (hi == 0) {
        const float r0 = sqrtf(fmaxf(d0, 0.0f));
        const float r1 = sqrtf(fmaxf(d1, 0.0f));
        const float r2 = sqrtf(fmaxf(d2, 0.0f));
        float w0 = 1.0f / (r0 + EPS_);
        float w1 = 1.0f / (r1 + EPS_);
        float w2 = 1.0f / (r2 + EPS_);
        const float inv = 1.0f / (w0 + w1 + w2);
        const int base = (b * N_ + n) * 3;
        knn_idx[base + 0] = i0;  knn_w[base + 0] = w0 * inv;
        knn_idx[base + 1] = i1;  knn_w[base + 1] = w1 * inv;
        knn_idx[base + 2] = i2;  knn_w[base + 2] = w2 * inv;
    }
}

// ---------------------------------------------------------------------------
// Phase 2: weighted feature gather (channels 0..C2-1) + tgt_feats passthrough.
// One thread per output element; consecutive threads cover consecutive n ->
// coalesced stores; src_feats rows (16 KB) are L2-resident for the 3-gather.
// ---------------------------------------------------------------------------
__global__ __launch_bounds__(256) void interp_concat_kernel(
    const float* __restrict__ src_feats,  // [B, C2, M]
    const float* __restrict__ tgt_feats,  // [B, C1, N]
    const int*   __restrict__ knn_idx,    // [B, N, 3]
    const float* __restrict__ knn_w,      // [B, N, 3]
    float* __restrict__ out)              // [B, C1+C2, N]
{
    const int e = blockIdx.x * 256 + threadIdx.x;   // total = B*CO*N
    const int n = e & (N_ - 1);
    const int c = (e >> 13) % CO_;                  // N_ == 1<<13
    const int b = e / (N_ * CO_);

    if (c < C2_) {
        const int base = (b * N_ + n) * 3;
        const int   j0 = knn_idx[base + 0];
        const int   j1 = knn_idx[base + 1];
        const int   j2 = knn_idx[base + 2];
        const float w0 = knn_w[base + 0];
        const float w1 = knn_w[base + 1];
        const float w2 = knn_w[base + 2];
        const float* row = src_feats + ((size_t)b * C2_ + c) * M_;
        __builtin_prefetch(row + j0, 0, 0);         // global_prefetch_b8
        out[e] = w0 * row[j0] + w1 * row[j1] + w2 * row[j2];
    } else {
        out[e] = tgt_feats[((size_t)b * C1_ + (c - C2_)) * N_ + n];
    }
}

// ---------------------------------------------------------------------------
extern "C" void kernel_launch(void* const* d_in, const int* in_sizes, int n_in,
                              void* d_out, int out_size, void* d_ws, size_t ws_size,
                              hipStream_t stream) {
    const float* tgt       = (const float*)d_in[0];  // (B, N, 3)
    const float* src       = (const float*)d_in[1];  // (B, M, 3)
    const float* tgt_feats = (const float*)d_in[2];  // (B, C1, N)
    const float* src_feats = (const float*)d_in[3];  // (B, C2, M)
    // d_in[4] is k (== 3); fixed at compile time.

    int*   knn_idx = (int*)d_ws;                                     // B*N*3 ints
    float* knn_w   = (float*)((char*)d_ws +
                              (size_t)B_ * N_ * 3 * sizeof(int));    // B*N*3 floats

    knn3_wmma_kernel<<<B_ * (N_ / 128), 256, 0, stream>>>(tgt, src, knn_idx, knn_w);

    const int total = B_ * CO_ * N_;
    interp_concat_kernel<<<total / 256, 256, 0, stream>>>(
        src_feats, tgt_feats, knn_idx, knn_w, (float*)d_out);
}